// Generator_7181185318928
// MI455X (gfx1250) — compile-verified
//
#include <hip/hip_runtime.h>
#include <hip/hip_bf16.h>
#include <math.h>

typedef __attribute__((ext_vector_type(2))) float v2f;
typedef __attribute__((ext_vector_type(8))) float v8f;

#define HIDDEN        128
#define NTILES        8      // 128 / 16 output-neuron tiles
#define KSTEPS        32     // 128 / 4  (K=4 per f32 WMMA)
#define THREADS       256
#define WAVES_PER_BLK 8
#define NBLOCKS       512
#define N_ROWS        524288

// ---------------------------------------------------------------------------
// Prep: fold the constant 13 input features into c1[128] (bias-like term) and
// extract w[128] (weight on the varying row index). One thread per neuron,
// threads 0..127 -> green MLP, 128..255 -> red MLP.
// ws layout: [0:128) c1_g, [128:256) w_g, [256:384) c1_r, [384:512) w_r
// ---------------------------------------------------------------------------
__global__ void prep_kernel(const float* __restrict__ z, const int* __restrict__ ld,
                            const float* __restrict__ gW1, const float* __restrict__ gb1,
                            const float* __restrict__ rW1, const float* __restrict__ rb1,
                            float* __restrict__ ws) {
    int tid = threadIdx.x;
    const float* W1; const float* b1; float* c1o; float* wo; int j;
    if (tid < 128) { W1 = gW1; b1 = gb1; c1o = ws;       wo = ws + 128; j = tid; }
    else           { W1 = rW1; b1 = rb1; c1o = ws + 256; wo = ws + 384; j = tid - 128; }
    float acc = b1[j];
#pragma unroll
    for (int k = 0; k < 10; ++k) acc += W1[j * 14 + k] * z[k];
    acc += W1[j * 14 + 10] * (float)ld[0];
    acc += W1[j * 14 + 11] * (float)ld[1];
    acc += W1[j * 14 + 12] * (float)ld[2];
    c1o[j] = acc;
    wo[j]  = W1[j * 14 + 13];
}

// ---------------------------------------------------------------------------
// Main kernel: one MLP instance. Each wave handles 16 rows per iteration of a
// persistent loop; layer-2 GEMM done with v_wmma_f32_16x16x4_f32 reading
// pre-swizzled B fragments of W2 from LDS.
// ---------------------------------------------------------------------------
__global__ __launch_bounds__(THREADS) void mlp_kernel(
    const float* __restrict__ c1w,  // c1[128] then w[128]
    const float* __restrict__ W2,   // [128][128] row-major (out, in)
    const float* __restrict__ b2,   // [128]
    const float* __restrict__ W3,   // [128]
    const float* __restrict__ b3p,  // [1]
    const int*   __restrict__ ld,   // level_data (grid size in [0])
    int*         __restrict__ out,  // [nrows] int32 coords
    int nrows)
{
    // B fragments of W2: [jt][t][lane] -> (W2[jt*16+nn][k0], W2[..][k0+1])
    // with nn = lane&15, k0 = 4t + 2*(lane>>4).  64 KB.
    __shared__ v2f   lw2[NTILES * KSTEPS * 32];
    __shared__ float lc1[HIDDEN], lw[HIDDEN], lb2[HIDDEN], lw3[HIDDEN];

    const int tid = threadIdx.x;

    for (int s = tid; s < NTILES * KSTEPS * 32; s += THREADS) {
        int jt = s >> 10;
        int t  = (s >> 5) & 31;
        int ln = s & 31;
        int nn = ln & 15;
        int k0 = 4 * t + ((ln >> 4) << 1);
        int j  = jt * 16 + nn;
        v2f v;
        v.x = W2[j * HIDDEN + k0];
        v.y = W2[j * HIDDEN + k0 + 1];
        lw2[s] = v;
    }
    if (tid < HIDDEN) {
        lc1[tid] = c1w[tid];
        lw[tid]  = c1w[HIDDEN + tid];
        lb2[tid] = b2[tid];
        lw3[tid] = W3[tid];
    }
    __syncthreads();

    const float b3    = b3p[0];
    const float g     = (float)ld[0];
    const float scale = g * g - 1.0f;

    const int lane  = tid & 31;
    const int wave  = tid >> 5;
    const int gwave = blockIdx.x * WAVES_PER_BLK + wave;
    const int m     = lane & 15;          // A-matrix row / C-matrix column
    const int khalf = (lane >> 4) << 1;   // K sub-offset per f32 A layout

    const int nblk16     = nrows >> 4;
    const int totalWaves = gridDim.x * WAVES_PER_BLK;

    for (int rb = gwave; rb < nblk16; rb += totalWaves) {
        const int   base = rb << 4;
        const float fi   = (float)(base + m);

        // Build A fragments: h1 = leaky_relu(c1 + w*i), laid out per ISA
        // 16x4 f32 A layout (VGPR0 = K {0|2}, VGPR1 = K {1|3}).
        v2f afrag[KSTEPS];
#pragma unroll
        for (int t = 0; t < KSTEPS; ++t) {
            int   k0 = 4 * t + khalf;
            float x0 = lc1[k0]     + lw[k0]     * fi;
            float x1 = lc1[k0 + 1] + lw[k0 + 1] * fi;
            afrag[t].x = x0 > 0.f ? x0 : 0.2f * x0;
            afrag[t].y = x1 > 0.f ? x1 : 0.2f * x1;
        }

        float p[8];
#pragma unroll
        for (int r = 0; r < 8; ++r) p[r] = 0.f;

        for (int jt = 0; jt < NTILES; ++jt) {
            v8f acc = {0.f, 0.f, 0.f, 0.f, 0.f, 0.f, 0.f, 0.f};
#pragma unroll
            for (int t = 0; t < KSTEPS; ++t) {
                v2f bfrag = lw2[(jt * KSTEPS + t) * 32 + lane];
                acc = __builtin_amdgcn_wmma_f32_16x16x4_f32(
                    false, afrag[t], false, bfrag, (short)0, acc, false, false);
            }
            // acc[r] = layer2 pre-activation for row M = r + 8*(lane>=16),
            // neuron n = jt*16 + (lane&15).  ELU then fold layer-3 dot.
            int   n  = jt * 16 + m;
            float bb = lb2[n];
            float w3 = lw3[n];
#pragma unroll
            for (int r = 0; r < 8; ++r) {
                float t2 = acc[r] + bb;
                float h2 = t2 > 0.f ? t2 : (__expf(t2) - 1.f);
                p[r] += h2 * w3;
            }
        }

        // Sum partial layer-3 products over the 16 lanes sharing each row.
#pragma unroll
        for (int r = 0; r < 8; ++r) {
            float v = p[r];
            v += __shfl_xor(v, 1, 32);
            v += __shfl_xor(v, 2, 32);
            v += __shfl_xor(v, 4, 32);
            v += __shfl_xor(v, 8, 32);
            p[r] = v;
        }

        if ((lane & 15) == 0) {
            int mo = (lane >> 4) * 8;
#pragma unroll
            for (int r = 0; r < 8; ++r) {
                float o  = p[r] + b3;
                float sg = 1.f / (1.f + __expf(-o * (1.0f / 500.0f)));
                out[base + mo + r] = (int)(sg * scale);
            }
        }
    }
}

// ---------------------------------------------------------------------------
extern "C" void kernel_launch(void* const* d_in, const int* in_sizes, int n_in,
                              void* d_out, int out_size, void* d_ws, size_t ws_size,
                              hipStream_t stream) {
    const float* z   = (const float*)d_in[0];
    const int*   ld  = (const int*)d_in[1];
    const float* gW1 = (const float*)d_in[2];
    const float* gb1 = (const float*)d_in[3];
    const float* gW2 = (const float*)d_in[4];
    const float* gb2 = (const float*)d_in[5];
    const float* gW3 = (const float*)d_in[6];
    const float* gb3 = (const float*)d_in[7];
    const float* rW1 = (const float*)d_in[8];
    const float* rb1 = (const float*)d_in[9];
    const float* rW2 = (const float*)d_in[10];
    const float* rb2 = (const float*)d_in[11];
    const float* rW3 = (const float*)d_in[12];
    const float* rb3 = (const float*)d_in[13];

    float* ws  = (float*)d_ws;
    int*   out = (int*)d_out;

    prep_kernel<<<1, 256, 0, stream>>>(z, ld, gW1, gb1, rW1, rb1, ws);
    mlp_kernel<<<NBLOCKS, THREADS, 0, stream>>>(ws,       gW2, gb2, gW3, gb3, ld, out,          N_ROWS);
    mlp_kernel<<<NBLOCKS, THREADS, 0, stream>>>(ws + 256, rW2, rb2, rW3, rb3, ld, out + N_ROWS, N_ROWS);
}